// GATNet_70755291234775
// MI455X (gfx1250) — compile-verified
//
#include <hip/hip_runtime.h>
#include <math.h>

// ---------------------------------------------------------------------------
// GAT network on gfx1250: fp32 WMMA (v_wmma_f32_16x16x4_f32) for the two
// feature GEMMs; atomic-based segment softmax / aggregation for edges.
// ---------------------------------------------------------------------------

#define LRELU_SLOPE 0.2f

typedef float v2f __attribute__((ext_vector_type(2)));
typedef float v8f __attribute__((ext_vector_type(8)));

// Monotone float <-> uint32 order-preserving encode (for float atomicMax).
__device__ __forceinline__ unsigned fenc(float f) {
    unsigned u = __float_as_uint(f);
    return (u & 0x80000000u) ? ~u : (u | 0x80000000u);
}
__device__ __forceinline__ float fdec(unsigned e) {
    unsigned u = (e & 0x80000000u) ? (e & 0x7FFFFFFFu) : ~e;
    return __uint_as_float(u);
}

// ---------------------------------------------------------------------------
// C[M x Nc] = A[M x K] @ B[K x Nc], fp32, via V_WMMA_F32_16X16X4_F32.
// One wave computes a 16x64 strip (4 accumulators, A fragment reused 4x).
// Requires: M % 16 == 0, Nc % 64 == 0, K % 4 == 0 (all hold here).
// ---------------------------------------------------------------------------
__global__ void wmma_gemm_f32(const float* __restrict__ A,
                              const float* __restrict__ B,
                              float* __restrict__ C,
                              int M, int K, int Nc)
{
    const int gtid = blockIdx.x * blockDim.x + threadIdx.x;
    const int wave = gtid >> 5;
    const int lane = threadIdx.x & 31;
    const int ntn  = Nc >> 6;              // 64-wide column strips
    const int mt   = wave / ntn;
    const int nt   = wave - mt * ntn;
    if (mt * 16 >= M) return;              // wave-uniform: EXEC stays all-1s
    const int m0 = mt * 16;
    const int n0 = nt * 64;
    const int half = lane >> 4;            // 0: K+{0,1}, 1: K+{2,3}
    const int l    = lane & 15;

    v8f acc0 = {}, acc1 = {}, acc2 = {}, acc3 = {};

    for (int k0 = 0; k0 < K; k0 += 4) {
        const int kb = k0 + 2 * half;
        // A fragment: lane l holds row m0+l, VGPR v holds K = kb + v
        const float* arow = A + (size_t)(m0 + l) * K + kb;
        v2f a; a.x = arow[0]; a.y = arow[1];
        // B fragments: lane l holds column n0+strip+l, VGPR v holds K = kb + v
        const float* b0 = B + (size_t)kb * Nc + n0 + l;
        const float* b1 = b0 + (size_t)Nc;
        v2f bb0, bb1, bb2, bb3;
        bb0.x = b0[0];  bb0.y = b1[0];
        bb1.x = b0[16]; bb1.y = b1[16];
        bb2.x = b0[32]; bb2.y = b1[32];
        bb3.x = b0[48]; bb3.y = b1[48];
        acc0 = __builtin_amdgcn_wmma_f32_16x16x4_f32(false, a, false, bb0, (short)0, acc0, false, false);
        acc1 = __builtin_amdgcn_wmma_f32_16x16x4_f32(false, a, false, bb1, (short)0, acc1, false, false);
        acc2 = __builtin_amdgcn_wmma_f32_16x16x4_f32(false, a, false, bb2, (short)0, acc2, false, false);
        acc3 = __builtin_amdgcn_wmma_f32_16x16x4_f32(false, a, false, bb3, (short)0, acc3, false, false);
    }

    // C/D layout: VGPR v, lane (half,l) -> row m0 + v + 8*half, col n0 + l
    for (int v = 0; v < 8; ++v) {
        float* crow = C + (size_t)(m0 + v + 8 * half) * Nc + n0 + l;
        crow[0]  = acc0[v];
        crow[16] = acc1[v];
        crow[32] = acc2[v];
        crow[48] = acc3[v];
    }
}

// ---------------------------------------------------------------------------
// Per-(node, head) attention scores: a_src/a_dst = <h[n,h,:], att[h,:]>.
// One wave per (node, head); wave32 shuffle reduction.
// ---------------------------------------------------------------------------
__global__ void att_scores(const float* __restrict__ h,
                           const float* __restrict__ att_s,
                           const float* __restrict__ att_d,
                           float* __restrict__ asrc,
                           float* __restrict__ adst,
                           int n, int heads, int C)
{
    const int wid  = (blockIdx.x * blockDim.x + threadIdx.x) >> 5;
    const int lane = threadIdx.x & 31;
    if (wid >= n * heads) return;
    const int node = wid / heads;
    const int hd   = wid - node * heads;
    const float* hp = h + (size_t)node * heads * C + (size_t)hd * C;
    const float* as = att_s + (size_t)hd * C;
    const float* ad = att_d + (size_t)hd * C;
    float s = 0.f, d = 0.f;
    for (int c = lane; c < C; c += 32) {
        float v = hp[c];
        s += v * as[c];
        d += v * ad[c];
    }
    for (int off = 16; off > 0; off >>= 1) {
        s += __shfl_down(s, off, 32);
        d += __shfl_down(d, off, 32);
    }
    if (lane == 0) { asrc[wid] = s; adst[wid] = d; }
}

__global__ void fill_u32(unsigned* __restrict__ p, unsigned v, int n)
{
    int i = blockIdx.x * blockDim.x + threadIdx.x;
    if (i < n) p[i] = v;
}

// edge id e in [0,E): src = ei[e], dst = ei[E+e];  e in [E, E+n): self loop.
__device__ __forceinline__ void edge_endpoints(const int* __restrict__ ei,
                                               int E, int e, int& s, int& d)
{
    if (e < E) { s = ei[e]; d = ei[E + e]; }
    else       { s = d = e - E; }
}

__device__ __forceinline__ float edge_alpha(const float* __restrict__ asrc,
                                            const float* __restrict__ adst,
                                            int s, int d, int heads, int hd)
{
    float a = asrc[s * heads + hd] + adst[d * heads + hd];
    return (a > 0.f) ? a : a * LRELU_SLOPE;
}

// Pass 1: segment max of leaky-relu'd scores into encoded-u32 buffer.
__global__ void edge_seg_max(const int* __restrict__ ei, int E, int n, int heads,
                             const float* __restrict__ asrc,
                             const float* __restrict__ adst,
                             unsigned* __restrict__ amax_enc)
{
    int idx = blockIdx.x * blockDim.x + threadIdx.x;
    int tot = (E + n) * heads;
    if (idx >= tot) return;
    int e = idx / heads, hd = idx - e * heads;
    int s, d; edge_endpoints(ei, E, e, s, d);
    float a = edge_alpha(asrc, adst, s, d, heads, hd);
    atomicMax(&amax_enc[d * heads + hd], fenc(a));
}

// Decode encoded max in place (u32 -> f32), with isfinite guard like reference.
__global__ void decode_max(unsigned* __restrict__ enc, int n)
{
    int i = blockIdx.x * blockDim.x + threadIdx.x;
    if (i >= n) return;
    float m = fdec(enc[i]);
    if (!isfinite(m)) m = 0.f;
    ((float*)enc)[i] = m;
}

// Pass 2: denom[dst,h] += exp(alpha - amax[dst,h])
__global__ void edge_seg_sum(const int* __restrict__ ei, int E, int n, int heads,
                             const float* __restrict__ asrc,
                             const float* __restrict__ adst,
                             const float* __restrict__ amax,
                             float* __restrict__ denom)
{
    int idx = blockIdx.x * blockDim.x + threadIdx.x;
    int tot = (E + n) * heads;
    if (idx >= tot) return;
    int e = idx / heads, hd = idx - e * heads;
    int s, d; edge_endpoints(ei, E, e, s, d);
    float a = edge_alpha(asrc, adst, s, d, heads, hd);
    atomicAdd(&denom[d * heads + hd], expf(a - amax[d * heads + hd]));
}

// Pass 3: agg[dst,h,:] += coef * h[src,h,:].  One wave per (edge, head).
__global__ void edge_aggregate(const int* __restrict__ ei, int E, int n,
                               int heads, int C,
                               const float* __restrict__ asrc,
                               const float* __restrict__ adst,
                               const float* __restrict__ amax,
                               const float* __restrict__ denom,
                               const float* __restrict__ h,
                               float* __restrict__ agg)
{
    const int wid  = (blockIdx.x * blockDim.x + threadIdx.x) >> 5;
    const int lane = threadIdx.x & 31;
    const int tot  = (E + n) * heads;
    if (wid >= tot) return;
    int e = wid / heads, hd = wid - e * heads;
    int s, d; edge_endpoints(ei, E, e, s, d);
    float a    = edge_alpha(asrc, adst, s, d, heads, hd);
    int   dh   = d * heads + hd;
    float coef = expf(a - amax[dh]) / fmaxf(denom[dh], 1e-16f);
    const float* hs = h   + (size_t)s * heads * C + (size_t)hd * C;
    float*       ag = agg + (size_t)d * heads * C + (size_t)hd * C;
    for (int c = lane; c < C; c += 32)
        atomicAdd(&ag[c], hs[c] * coef);
}

// in-place ELU(x + bias)
__global__ void elu_bias(float* __restrict__ a, const float* __restrict__ b,
                         int n, int cols)
{
    int i = blockIdx.x * blockDim.x + threadIdx.x;
    if (i >= n) return;
    float v = a[i] + b[i % cols];
    a[i] = (v > 0.f) ? v : expm1f(v);
}

// relu(agg + b2) then per-graph max-pool (values >= 0 so raw-bit atomicMax on a
// zero-initialized accumulator matches segment_max + where(isfinite,...,0)).
__global__ void pool_relu_max(const float* __restrict__ agg,
                              const float* __restrict__ b2,
                              const int* __restrict__ batch,
                              unsigned* __restrict__ pooled,
                              int n, int C)
{
    int i = blockIdx.x * blockDim.x + threadIdx.x;
    if (i >= n * C) return;
    int node = i / C, c = i - node * C;
    float v = fmaxf(agg[i] + b2[c], 0.f);
    atomicMax(&pooled[batch[node] * C + c], __float_as_uint(v));
}

// FC head: [64,128] -> relu([64,16]) -> [64,1].  Single 1024-thread block.
__global__ void fc_head(const float* __restrict__ pooled,
                        const float* __restrict__ w1, const float* __restrict__ b1,
                        const float* __restrict__ w2, const float* __restrict__ b2,
                        float* __restrict__ out)
{
    __shared__ float t[64 * 16];
    const int tid = threadIdx.x;        // 0..1023
    const int g = tid >> 4, j = tid & 15;
    float s = b1[j];
    const float* p = pooled + (size_t)g * 128;
    for (int c = 0; c < 128; ++c) s += p[c] * w1[c * 16 + j];
    t[tid] = fmaxf(s, 0.f);
    __syncthreads();
    if (j == 0) {
        float o = b2[0];
        for (int k = 0; k < 16; ++k) o += t[(g << 4) + k] * w2[k];
        out[g] = o;
    }
}

// ---------------------------------------------------------------------------
static inline int cdiv(long long a, long long b) { return (int)((a + b - 1) / b); }

extern "C" void kernel_launch(void* const* d_in, const int* in_sizes, int n_in,
                              void* d_out, int out_size, void* d_ws, size_t ws_size,
                              hipStream_t stream)
{
    const float* x        = (const float*)d_in[0];
    const int*   ei       = (const int*)  d_in[1];   // [2,E] flat: [0..E)=src, [E..2E)=dst
    const int*   batch    = (const int*)  d_in[2];
    const float* W1       = (const float*)d_in[3];   // [128, 512]
    const float* att1_src = (const float*)d_in[4];   // [4,128]
    const float* att1_dst = (const float*)d_in[5];
    const float* b1       = (const float*)d_in[6];   // [512]
    const float* W2       = (const float*)d_in[7];   // [512, 128]
    const float* att2_src = (const float*)d_in[8];   // [1,128]
    const float* att2_dst = (const float*)d_in[9];
    const float* b2       = (const float*)d_in[10];  // [128]
    const float* fc1_w    = (const float*)d_in[11];  // [128,16]
    const float* fc1_b    = (const float*)d_in[12];  // [16]
    const float* fc2_w    = (const float*)d_in[13];  // [16,1]
    const float* fc2_b    = (const float*)d_in[14];  // [1]
    float*       out      = (float*)d_out;           // [64]

    const int N = in_sizes[2];          // 20000
    const int E = in_sizes[1] / 2;      // 320000
    const int D = 128, H1 = 4, C = 128, G = 64;
    const int ET = E + N;               // edges incl. self loops

    // Workspace layout (fp32 elements)
    float* h1    = (float*)d_ws;                    // N*512
    float* agg1  = h1   + (size_t)N * (H1 * C);     // N*512 (later: hin after ELU)
    float* asrc  = agg1 + (size_t)N * (H1 * C);     // N*4
    float* adst  = asrc + (size_t)N * H1;           // N*4
    float* amax  = adst + (size_t)N * H1;           // N*4 (u32-encoded during max pass)
    float* denom = amax + (size_t)N * H1;           // N*4
    float* h2    = h1;                              // N*128 (reuse h1 region)
    float* agg2  = h1 + (size_t)N * C;              // N*128
    float* pooled= h1 + (size_t)N * C * 2;          // G*128
    (void)ws_size;

    const int TB = 256;
    const unsigned ENC_NEGINF = 0x007FFFFFu;        // fenc(-inf)

    // ---------------- Layer 1 ----------------
    {
        const int K = D, Nc = H1 * C;               // 128 -> 512
        int waves = (N / 16) * (Nc / 64);
        wmma_gemm_f32<<<cdiv((long long)waves * 32, TB), TB, 0, stream>>>(x, W1, h1, N, K, Nc);

        int sw = N * H1;                            // (node, head) waves
        att_scores<<<cdiv((long long)sw * 32, TB), TB, 0, stream>>>(h1, att1_src, att1_dst, asrc, adst, N, H1, C);

        fill_u32<<<cdiv(N * H1, TB), TB, 0, stream>>>((unsigned*)amax, ENC_NEGINF, N * H1);
        fill_u32<<<cdiv(N * H1, TB), TB, 0, stream>>>((unsigned*)denom, 0u, N * H1);
        fill_u32<<<cdiv((long long)N * H1 * C, TB), TB, 0, stream>>>((unsigned*)agg1, 0u, N * H1 * C);

        int et = ET * H1;
        edge_seg_max<<<cdiv(et, TB), TB, 0, stream>>>(ei, E, N, H1, asrc, adst, (unsigned*)amax);
        decode_max<<<cdiv(N * H1, TB), TB, 0, stream>>>((unsigned*)amax, N * H1);
        edge_seg_sum<<<cdiv(et, TB), TB, 0, stream>>>(ei, E, N, H1, asrc, adst, amax, denom);
        edge_aggregate<<<cdiv((long long)et * 32, TB), TB, 0, stream>>>(ei, E, N, H1, C, asrc, adst, amax, denom, h1, agg1);

        elu_bias<<<cdiv((long long)N * H1 * C, TB), TB, 0, stream>>>(agg1, b1, N * H1 * C, H1 * C);
    }

    // ---------------- Layer 2 ----------------
    {
        const int K = H1 * C, Nc = C;               // 512 -> 128, heads=1
        int waves = (N / 16) * (Nc / 64);
        wmma_gemm_f32<<<cdiv((long long)waves * 32, TB), TB, 0, stream>>>(agg1, W2, h2, N, K, Nc);

        att_scores<<<cdiv((long long)N * 32, TB), TB, 0, stream>>>(h2, att2_src, att2_dst, asrc, adst, N, 1, C);

        fill_u32<<<cdiv(N, TB), TB, 0, stream>>>((unsigned*)amax, ENC_NEGINF, N);
        fill_u32<<<cdiv(N, TB), TB, 0, stream>>>((unsigned*)denom, 0u, N);
        fill_u32<<<cdiv((long long)N * C, TB), TB, 0, stream>>>((unsigned*)agg2, 0u, N * C);
        fill_u32<<<cdiv(G * C, TB), TB, 0, stream>>>((unsigned*)pooled, 0u, G * C);

        edge_seg_max<<<cdiv(ET, TB), TB, 0, stream>>>(ei, E, N, 1, asrc, adst, (unsigned*)amax);
        decode_max<<<cdiv(N, TB), TB, 0, stream>>>((unsigned*)amax, N);
        edge_seg_sum<<<cdiv(ET, TB), TB, 0, stream>>>(ei, E, N, 1, asrc, adst, amax, denom);
        edge_aggregate<<<cdiv((long long)ET * 32, TB), TB, 0, stream>>>(ei, E, N, 1, C, asrc, adst, amax, denom, h2, agg2);

        pool_relu_max<<<cdiv((long long)N * C, TB), TB, 0, stream>>>(agg2, b2, batch, (unsigned*)pooled, N, C);
    }

    // ---------------- FC head ----------------
    fc_head<<<1, 1024, 0, stream>>>(pooled, fc1_w, fc1_b, fc2_w, fc2_b, out);
    (void)out_size; (void)n_in;
}